// GCNLayer_40415642255629
// MI455X (gfx1250) — compile-verified
//
#include <hip/hip_runtime.h>

// GCN layer on MI455X (gfx1250), memory-bound on streaming adj (1 GiB) twice:
//   pass 1: deg = colsum(adj)+1 -> dinv = rsqrt(deg)
//   pass 2: out = relu(dinv*(adj^T @ z + z) + b),  z = dinv*(x@W)
// ~92us HBM floor @ 23.3 TB/s; compute (34 GFLOP) is free even on f32 WMMA.
// adj^T @ z uses V_WMMA_F32_16X16X4_F32 so tiles need NO conversion ->
// staged straight into LDS by the Tensor Data Mover (tensor_load_to_lds),
// double-buffered under TENSORcnt, fully overlapping copy with WMMA compute.

#define NN 16384
#define FD 64
#define KB 32     // k-block per TDM staging round (8 WMMA k-steps of 4)
#define TI 128    // output rows (adj columns) per workgroup

typedef __attribute__((ext_vector_type(2))) float v2f;
typedef __attribute__((ext_vector_type(8))) float v8f;
typedef unsigned int u32x4 __attribute__((ext_vector_type(4)));
typedef int          i32x4 __attribute__((ext_vector_type(4)));
typedef int          i32x8 __attribute__((ext_vector_type(8)));

// ---- TDM: issue one 2D tile load (global -> LDS), D# per ISA cdna5 §8 ----
__device__ __forceinline__ void tdm_load_2d(unsigned lds_off, const void* gptr,
                                            unsigned tile_w, unsigned tile_h,
                                            unsigned stride_elts,
                                            unsigned tensor_w, unsigned tensor_h) {
  unsigned long long ga = (unsigned long long)(uintptr_t)gptr;
  u32x4 g0;
  g0[0] = 1u;                                  // count=1, user-mode
  g0[1] = lds_off;                             // lds_addr (bytes)
  g0[2] = (unsigned)ga;                        // global_addr[31:0]
  g0[3] = (unsigned)(ga >> 32) | (2u << 30);   // global_addr[56:32] | type=2
  i32x8 g1;
  g1[0] = (int)(2u << 16);                     // data_size=2 (4B); no multicast/pad
  g1[1] = (int)((tensor_w & 0xFFFFu) << 16);   // tensor_dim0[15:0] @ bit48
  g1[2] = (int)(((tensor_w >> 16) & 0xFFFFu) | ((tensor_h & 0xFFFFu) << 16));
  g1[3] = (int)(((tensor_h >> 16) & 0xFFFFu) | (tile_w << 16));   // tile_dim0
  g1[4] = (int)tile_h;                         // tile_dim1 (tile_dim2=0)
  g1[5] = (int)stride_elts;                    // tensor_dim0_stride[31:0]
  g1[6] = 0;
  g1[7] = 0;
  i32x4 z4 = {0, 0, 0, 0};                     // groups 2/3 unused (<=2D tensor)
  i32x8 z8 = {0, 0, 0, 0, 0, 0, 0, 0};
  // clang-23 / therock-10.0 form: 6 args (g0, g1, g2, g3, extra group, cpol)
  __builtin_amdgcn_tensor_load_to_lds(g0, g1, z4, z4, z8, 0);
}

__device__ __forceinline__ unsigned lds_byte_off(const void* p) {
  // addrspace(3) -> generic puts the LDS byte offset in the low 32 bits
  return (unsigned)(unsigned long long)(uintptr_t)p;
}

// ---------------- phase 1: column degree ----------------
__global__ void zero_deg_kernel(float* __restrict__ deg) {
  int i = blockIdx.x * blockDim.x + threadIdx.x;
  if (i < NN) deg[i] = 0.0f;
}

__global__ void colsum_kernel(const float* __restrict__ adj,
                              float* __restrict__ deg) {
  int col = blockIdx.x * 256 + threadIdx.x;
  int r0  = blockIdx.y * 256;
  float s = 0.0f;
  #pragma unroll 4
  for (int r = 0; r < 256; ++r)
    s += adj[(size_t)(r0 + r) * NN + col];
  atomicAdd(&deg[col], s);   // global_atomic_add_f32
}

__global__ void dinv_kernel(const float* __restrict__ deg,
                            float* __restrict__ dinv) {
  int i = blockIdx.x * blockDim.x + threadIdx.x;
  if (i < NN) dinv[i] = rsqrtf(deg[i] + 1.0f);   // deg+1 >= 1 always
}

// ---------------- phase 2: z = dinv * (x @ W) ----------------
__global__ void __launch_bounds__(256)
xw_kernel(const float* __restrict__ x, const float* __restrict__ W,
          const float* __restrict__ dinv, float* __restrict__ zf) {
  __shared__ float sW[FD * FD];
  for (int t = threadIdx.x; t < FD * FD; t += blockDim.x) sW[t] = W[t];
  __syncthreads();
  int idx = blockIdx.x * blockDim.x + threadIdx.x;   // one (i,f)
  int i = idx >> 6;
  int f = idx & 63;
  if (i < NN) {
    const float* xr = &x[(size_t)i * FD];
    float s = 0.0f;
    #pragma unroll
    for (int k = 0; k < FD; ++k) s = fmaf(xr[k], sW[k * FD + f], s);
    zf[idx] = dinv[i] * s;
  }
}

// -------- phase 3: out = relu(dinv*(adj^T @ z + z) + b), TDM + f32 WMMA --------
__global__ void __launch_bounds__(256)
spmm_wmma_kernel(const float* __restrict__ adj,
                 const float* __restrict__ zf,
                 const float* __restrict__ dinv,
                 const float* __restrict__ bias,
                 float* __restrict__ out) {
  __shared__ float sA[2][KB][TI];   // adj tile, [k][i], 2 x 16 KB
  __shared__ float sZ[2][KB][FD];   // z   tile, [k][f], 2 x  8 KB

  const int tid   = threadIdx.x;
  const int lane  = tid & 31;       // wave32
  const int wv    = tid >> 5;       // 8 waves
  const int li    = lane & 15;
  const int hi    = lane >> 4;
  const int i0    = blockIdx.x * TI;
  const int iBase = wv * 16;

  v8f acc[4] = {};                  // 16(i) x 64(f) f32 accumulators per wave

  const int R = NN / KB;            // 512 rounds

  // Prologue: TDM-stage round 0 into buffer 0 (wave 0 drives the DMA engine).
  if (wv == 0) {
    tdm_load_2d(lds_byte_off(&sA[0][0][0]), adj + (size_t)0 * NN + i0,
                TI, KB, NN, NN, NN);
    tdm_load_2d(lds_byte_off(&sZ[0][0][0]), zf,
                FD, KB, FD, FD, NN);
  }

  for (int r = 0; r < R; ++r) {
    const int buf = r & 1;
    const int kb  = r * KB;

    if (wv == 0) {
      if (r + 1 < R) {
        // Kick off next round's tiles, then wait only for the current pair:
        // the new pair streams in while all waves run WMMA below.
        tdm_load_2d(lds_byte_off(&sA[buf ^ 1][0][0]),
                    adj + (size_t)(kb + KB) * NN + i0, TI, KB, NN, NN, NN);
        tdm_load_2d(lds_byte_off(&sZ[buf ^ 1][0][0]),
                    zf + (size_t)(kb + KB) * FD, FD, KB, FD, FD, NN);
        __builtin_amdgcn_s_wait_tensorcnt(2);
      } else {
        __builtin_amdgcn_s_wait_tensorcnt(0);
      }
    }
    __syncthreads();                // publish current buffer to all waves

    // 8 k-steps of 4 over this block. f32 WMMA fragment layouts (ISA 7.12.2):
    //   A 16x4: lane holds row M=li, VGPR0/1 = K = 2*hi, 2*hi+1
    //   B 4x16: lane holds col N=li, VGPR0/1 = K = 2*hi, 2*hi+1
    #pragma unroll
    for (int ks = 0; ks < KB; ks += 4) {
      v2f a;
      a[0] = sA[buf][ks + 2 * hi + 0][iBase + li];
      a[1] = sA[buf][ks + 2 * hi + 1][iBase + li];
      #pragma unroll
      for (int ft = 0; ft < 4; ++ft) {
        v2f b;
        b[0] = sZ[buf][ks + 2 * hi + 0][ft * 16 + li];
        b[1] = sZ[buf][ks + 2 * hi + 1][ft * 16 + li];
        acc[ft] = __builtin_amdgcn_wmma_f32_16x16x4_f32(
            false, a, false, b, (short)0, acc[ft], false, false);
      }
    }
    __syncthreads();                // all reads done before next TDM overwrite
  }

  // Epilogue: D layout VGPR r -> M = r + 8*hi, N = li. Fuse self-loop,
  // dinv scaling, bias, relu.
  #pragma unroll
  for (int ft = 0; ft < 4; ++ft) {
    #pragma unroll
    for (int rr = 0; rr < 8; ++rr) {
      int gi = i0 + iBase + rr + hi * 8;
      int gf = ft * 16 + li;
      float v   = acc[ft][rr];
      float res = fmaf(dinv[gi], v + zf[(size_t)gi * FD + gf], bias[gf]);
      out[(size_t)gi * FD + gf] = fmaxf(res, 0.0f);
    }
  }
}

extern "C" void kernel_launch(void* const* d_in, const int* in_sizes, int n_in,
                              void* d_out, int out_size, void* d_ws, size_t ws_size,
                              hipStream_t stream) {
  (void)in_sizes; (void)n_in; (void)out_size; (void)ws_size;
  const float* x   = (const float*)d_in[0];
  const float* adj = (const float*)d_in[1];
  const float* W   = (const float*)d_in[2];
  const float* b   = (const float*)d_in[3];
  float* out = (float*)d_out;

  // Workspace: deg (64 KB) | dinv (64 KB) | z_f32 (4 MB)
  char* ws = (char*)d_ws;
  float* deg  = (float*)ws;
  float* dinv = (float*)(ws + (64u << 10));
  float* zf   = (float*)(ws + (128u << 10));

  zero_deg_kernel<<<NN / 256, 256, 0, stream>>>(deg);
  {
    dim3 grid(NN / 256, NN / 256);
    colsum_kernel<<<grid, 256, 0, stream>>>(adj, deg);
  }
  dinv_kernel<<<NN / 256, 256, 0, stream>>>(deg, dinv);
  xw_kernel<<<(NN * FD) / 256, 256, 0, stream>>>(x, W, dinv, zf);
  spmm_wmma_kernel<<<NN / TI, 256, 0, stream>>>(adj, zf, dinv, b, out);
}